// MT2VEncoderFusion_17592186045121
// MI455X (gfx1250) — compile-verified
//
#include <hip/hip_runtime.h>

// ---------------------------------------------------------------------------
// MT2VEncoderFusion for MI455X (gfx1250, wave32)
//
// out[b,d,c,i,j] = method_{argmax(w[b,d,:])}(xn[b,d,i], xn[b,d,j]),  c=0..2
// where xn is the resampled (512->224), min-max normalized, clipped series.
//
// Roofline: output = 16*8*3*224*224*4B = 77 MB, inputs ~260 KB -> pure
// store-BW bound; floor = 77MB / 23.3 TB/s ~= 3.3 us. Per-pixel compute is
// ~free: the trig methods are rank-2 outer products using
// s = sin(arccos(x)) = sqrt(1-x^2):
//   gasf = x_i*x_j - s_i*s_j      (A=[x,-s], B=[x,s]^T)
//   gadf = s_i*x_j - x_i*s_j      (A=[s,-x], B=[x,s]^T)
//   prod = x_i*x_j                (rank-1)
// -> one V_WMMA_F32_16X16X4_F32 per 16x16 tile (fp32, full precision).
// rp = 1-|x_i-x_j| is not low-rank -> VALU in the same C/D layout.
//
// Store path: WMMA C/D layout is column-per-lane (bad for packing), so each
// wave stages its tile row-major in a private 1KB LDS slice (8 ds_store_b32
// per lane) and drains it with GLOBAL_STORE_ASYNC_FROM_LDS_B128: 6 async
// wave-instructions per tile (512B each) instead of 24 b32 stores -> 4x
// fewer store issues at the same 64B/row DRAM contiguity. ASYNCcnt tracks
// completion; DS->async ordering enforced with s_wait_dscnt.
// ---------------------------------------------------------------------------

#define B_DIM 16
#define D_DIM 8
#define BD    (B_DIM * D_DIM)      // 128 series
#define L_IN  512
#define S_IMG 224
#define TILES_1D (S_IMG / 16)      // 14
#define TILES_PER_IMG (TILES_1D * TILES_1D)  // 196
#define IMG_PIX (S_IMG * S_IMG)    // 50176
#define N_TILES (BD * TILES_PER_IMG)         // 25088 (divisible by 8 waves)

typedef __attribute__((ext_vector_type(2))) float v2f;
typedef __attribute__((ext_vector_type(8))) float v8f;

// ---------------------------------------------------------------------------
// Kernel 1: resample + normalize + s=sqrt(1-x^2) + per-series method argmax.
// One block per (b,d) series; 256 threads (8 wave32), 224 active for samples.
// ---------------------------------------------------------------------------
__global__ __launch_bounds__(256) void
mt2v_prep_kernel(const float* __restrict__ x,        // [16,512,8]
                 const float* __restrict__ w,        // [16,8,4]
                 float* __restrict__ xn,             // [128,224]
                 float* __restrict__ sn,             // [128,224]
                 int*   __restrict__ midx)           // [128]
{
    const int bd = blockIdx.x;            // 0..127
    const int b  = bd >> 3;
    const int d  = bd & 7;
    const int t  = threadIdx.x;

    __shared__ float smn[256];
    __shared__ float smx[256];

    float v = 0.0f;
    if (t < S_IMG) {
        // pos = linspace(0, 511, 224)[t]
        const float pos  = (float)t * (511.0f / 223.0f);
        int i0 = (int)floorf(pos);
        i0 = i0 > (L_IN - 1) ? (L_IN - 1) : (i0 < 0 ? 0 : i0);
        const int i1 = i0 + 1 > (L_IN - 1) ? (L_IN - 1) : i0 + 1;
        const float frac = pos - (float)i0;
        const float* xb = x + (size_t)b * (L_IN * D_DIM) + d;
        const float a0 = xb[(size_t)i0 * D_DIM];
        const float a1 = xb[(size_t)i1 * D_DIM];
        v = a0 * (1.0f - frac) + a1 * frac;
        smn[t] = v;
        smx[t] = v;
    } else {
        smn[t] =  3.0e38f;
        smx[t] = -3.0e38f;
    }
    __syncthreads();

    // tree reduction for min / max over the 224 samples
    for (int s = 128; s > 0; s >>= 1) {
        if (t < s) {
            smn[t] = fminf(smn[t], smn[t + s]);
            smx[t] = fmaxf(smx[t], smx[t + s]);
        }
        __syncthreads();
    }
    const float mn = smn[0];
    const float mx = smx[0];

    if (t < S_IMG) {
        const float range = mx - mn + 1e-8f;
        float xv = 2.0f * (v - mn) / range - 1.0f;
        const float clip = 1.0f - 1e-6f;
        xv = fminf(fmaxf(xv, -clip), clip);
        xn[bd * S_IMG + t] = xv;
        sn[bd * S_IMG + t] = sqrtf(fmaxf(1.0f - xv * xv, 0.0f)); // sin(arccos) >= 0
    }
    if (t == 0) {
        const float* wp = w + bd * 4;
        int   best = 0;
        float bv   = wp[0];
        #pragma unroll
        for (int k = 1; k < 4; ++k) {
            if (wp[k] > bv) { bv = wp[k]; best = k; }   // first-max, like jnp.argmax
        }
        midx[bd] = best;
    }
}

// ---------------------------------------------------------------------------
// Kernel 2: one wave32 per 16x16 output tile.
// ---------------------------------------------------------------------------
__global__ __launch_bounds__(256) void
mt2v_img_kernel(const float* __restrict__ xn,   // [128,224]
                const float* __restrict__ sn,   // [128,224]
                const int*   __restrict__ midx, // [128]
                float* __restrict__ out)        // [16,8,3,224,224]
{
    const int wave = (blockIdx.x * blockDim.x + threadIdx.x) >> 5; // global tile id
    const int lane = threadIdx.x & 31;
    const int wv   = threadIdx.x >> 5;                             // wave in block

    const int bd = wave / TILES_PER_IMG;                 // 0..127 (wave-uniform)
    const int tt = wave - bd * TILES_PER_IMG;            // 0..195
    const int ti = tt / TILES_1D;
    const int tj = tt - ti * TILES_1D;
    const int rowbase = ti * 16;
    const int colbase = tj * 16;

    const float* __restrict__ xr = xn + bd * S_IMG;
    const float* __restrict__ sr = sn + bd * S_IMG;

    // method is uniform across the wave -> force scalar so the branch is an
    // s_cbranch and EXEC is all-ones at the WMMA (ISA requirement).
    const int method = __builtin_amdgcn_readfirstlane(midx[bd]);

    const int hi = lane >> 4;       // 0: lanes 0-15, 1: lanes 16-31
    const int m  = lane & 15;

    v8f acc;
    if (method == 2) {
        // rp = 1 - |x_i - x_j|  (VALU, C/D layout)
        const float xc = xr[colbase + m];
        #pragma unroll
        for (int v = 0; v < 8; ++v) {
            const int mr = v + (hi ? 8 : 0);
            const float xi = xr[rowbase + mr];
            acc[v] = 1.0f - fabsf(xi - xc);
        }
    } else {
        // Rank-2 outer product via fp32 WMMA 16x16x4.
        // A (16x4): lane<16 holds row m, K={0,1}; lane>=16 holds K={2,3}=0.
        // B (4x16): lane<16 holds col n=m, K={0,1}; lane>=16 K={2,3}=0.
        v2f a;    a.x = 0.0f;    a.y = 0.0f;
        v2f bm;   bm.x = 0.0f;   bm.y = 0.0f;
        if (hi == 0) {
            const float xi = xr[rowbase + m];
            const float si = sr[rowbase + m];
            if (method == 0)      { a.x = xi; a.y = -si; }  // gasf = x*x' - s*s'
            else if (method == 1) { a.x = si; a.y = -xi; }  // gadf = s*x' - x*s'
            else                  { a.x = xi; a.y = 0.0f; } // prod = x*x'
            bm.x = xr[colbase + m];
            bm.y = sr[colbase + m];
        }
        v8f c = {};
        acc = __builtin_amdgcn_wmma_f32_16x16x4_f32(
            /*neg_a=*/false, a, /*neg_b=*/false, bm,
            /*c_mod=*/(short)0, c, /*reuse_a=*/false, /*reuse_b=*/false);
    }

    // -------- Stage tile row-major in this wave's private LDS slice --------
    // C/D layout: lane l, VGPR v -> row mr = v + (l>=16 ? 8 : 0), col n = l&15
    __shared__ float tile[8][256];     // 1KB per wave, 8KB per block
    float* tp = &tile[wv][0];
    #pragma unroll
    for (int v = 0; v < 8; ++v) {
        const int mr = v + (hi ? 8 : 0);
        tp[mr * 16 + m] = acc[v];
    }
    // Async ops are unordered vs DS ops: make LDS writes visible first.
    asm volatile("s_wait_dscnt 0x0" ::: "memory");

    // -------- Drain LDS -> global with async B128 stores (ASYNCcnt) -------
    // Lane l moves 16B: rows lane/4 (chunk 0) and 8+lane/4 (chunk 1),
    // cols colbase + (lane%4)*4 .. +3  (64B contiguous per image row).
    const int r  = lane >> 2;           // 0..7
    const int c4 = (lane & 3) << 2;     // 0,4,8,12
    // Low 32 bits of a generic pointer to LDS == wave-relative LDS byte addr.
    const unsigned lds0 = (unsigned)(uintptr_t)(tp + lane * 4);
    const unsigned lds1 = lds0 + 512u;  // +128 floats (rows 8..15)

    const float* gb = out + (size_t)bd * 3 * IMG_PIX;
    #pragma unroll
    for (int ch = 0; ch < 3; ++ch) {
        const float* g0 = gb + (size_t)ch * IMG_PIX
                             + (size_t)(rowbase + r) * S_IMG + colbase + c4;
        const float* g1 = g0 + 8 * S_IMG;
        asm volatile("global_store_async_from_lds_b128 %0, %1, off"
                     :: "v"(g0), "v"(lds0) : "memory");
        asm volatile("global_store_async_from_lds_b128 %0, %1, off"
                     :: "v"(g1), "v"(lds1) : "memory");
    }
    // Ensure all async LDS->memory transfers completed before wave exit.
    asm volatile("s_wait_asynccnt 0x0" ::: "memory");
}

// ---------------------------------------------------------------------------
extern "C" void kernel_launch(void* const* d_in, const int* in_sizes, int n_in,
                              void* d_out, int out_size, void* d_ws, size_t ws_size,
                              hipStream_t stream) {
    const float* x = (const float*)d_in[0];   // [16,512,8]  fp32
    const float* w = (const float*)d_in[1];   // [16,8,4]    fp32
    float* out     = (float*)d_out;           // [16,8,3,224,224] fp32

    // workspace layout (256B-aligned sections)
    char* ws = (char*)d_ws;
    float* xn  = (float*)(ws);                 // 128*224 floats = 114688 B
    float* sn  = (float*)(ws + 114688);        // 128*224 floats = 114688 B
    int*   idx = (int*)  (ws + 229376);        // 128 ints

    mt2v_prep_kernel<<<BD, 256, 0, stream>>>(x, w, xn, sn, idx);
    mt2v_img_kernel<<<N_TILES / 8, 256, 0, stream>>>(xn, sn, idx, out);
}